// GATLayer_91079076479392
// MI455X (gfx1250) — compile-verified
//
#include <hip/hip_runtime.h>

#define BB 4
#define NN 2048
#define DIN 128
#define DOUT 128
#define NH 4
#define NEG_SLOPE 0.2f

typedef __attribute__((ext_vector_type(2))) float v2f;
typedef __attribute__((ext_vector_type(8))) float v8f;

// D = A(16x4) * B(4x16) + C, fp32 exact, wave32.
// 8-arg pattern: (neg_a, A, neg_b, B, c_mod, C, reuse_a, reuse_b)
__device__ __forceinline__ v8f wmma4(v2f a, v2f b, v8f c) {
  return __builtin_amdgcn_wmma_f32_16x16x4_f32(false, a, false, b, (short)0, c,
                                               false, false);
}

// ---------------------------------------------------------------------------
// 32x32 C tile per wave (2x2 register blocking over 16x16 WMMA tiles).
// A: row-major [.., lda], rows r0..r0+31 (pre-offset); lanes 0-15 and 16-31
//    both address the same 16 rows, K split {k0,k0+1}/{k0+2,k0+3} per ISA.
// B: row-major [K, ldb], cols c0..c0+31 (pre-offset).
// Per 4 WMMAs: 2 b64 + 4 b32 loads = 1.5 VMEM/WMMA.
// ---------------------------------------------------------------------------
template <int K>
__device__ __forceinline__ void gemm_tile32(const float* __restrict__ A,
                                            size_t lda,
                                            const float* __restrict__ Bm,
                                            size_t ldb, v8f acc[2][2], int half,
                                            int l16) {
  const float* Ar0 = A + (size_t)l16 * lda;         // rows 0..15 of tile
  const float* Ar1 = A + (size_t)(l16 + 16) * lda;  // rows 16..31 of tile
  for (int k0 = 0; k0 < K; k0 += 4) {
    const int kb = k0 + 2 * half;
    v2f a0 = *(const v2f*)(Ar0 + kb);
    v2f a1 = *(const v2f*)(Ar1 + kb);
    v2f b0, b1;
    b0.x = Bm[(size_t)kb * ldb + l16];
    b0.y = Bm[(size_t)(kb + 1) * ldb + l16];
    b1.x = Bm[(size_t)kb * ldb + 16 + l16];
    b1.y = Bm[(size_t)(kb + 1) * ldb + 16 + l16];
    acc[0][0] = wmma4(a0, b0, acc[0][0]);
    acc[0][1] = wmma4(a0, b1, acc[0][1]);
    acc[1][0] = wmma4(a1, b0, acc[1][0]);
    acc[1][1] = wmma4(a1, b1, acc[1][1]);
  }
}

// ---------------------------------------------------------------------------
// Stage 1: h[b,hd,n,o] = sum_i X[b,n,i] * W[hd,i,o]
// grid (NN/32, NH, BB), block 128 (4 waves); wave w -> cols c0=w*32
// ---------------------------------------------------------------------------
__global__ void __launch_bounds__(128)
k_proj(const float* __restrict__ X, const float* __restrict__ W,
       float* __restrict__ h) {
  const int tm = blockIdx.x, hd = blockIdx.y, b = blockIdx.z;
  const int wave = threadIdx.x >> 5;
  const int lane = threadIdx.x & 31;
  const int half = lane >> 4;
  const int l16  = lane & 15;
  const int r0 = tm * 32;
  const int c0 = wave * 32;

  const float* A  = X + ((size_t)b * NN + r0) * DIN;
  const float* Bm = W + (size_t)hd * DIN * DOUT + c0;
  v8f acc[2][2] = {};
  gemm_tile32<DIN>(A, DIN, Bm, DOUT, acc, half, l16);

  float* hp = h + ((size_t)b * NH + hd) * NN * DOUT;
#pragma unroll
  for (int i = 0; i < 2; ++i)
#pragma unroll
    for (int j = 0; j < 2; ++j)
#pragma unroll
      for (int v = 0; v < 8; ++v) {
        const int row = r0 + 16 * i + v + 8 * half;
        hp[(size_t)row * DOUT + c0 + 16 * j + l16] = acc[i][j][v];
      }
}

// ---------------------------------------------------------------------------
// Stage 2: s[r] = h[r,:]·a_src, d[r] = h[r,:]·a_dst ; one block per row
// ---------------------------------------------------------------------------
__global__ void __launch_bounds__(128)
k_score(const float* __restrict__ h, const float* __restrict__ a,
        float* __restrict__ sbuf, float* __restrict__ dbuf) {
  __shared__ float red0[128];
  __shared__ float red1[128];
  const int row = blockIdx.x;               // 0 .. B*H*N-1
  const int hd  = (row >> 11) & (NH - 1);   // N = 2048
  const int t   = threadIdx.x;
  const float hv = h[(size_t)row * DOUT + t];
  red0[t] = hv * a[hd * 2 * DOUT + t];
  red1[t] = hv * a[hd * 2 * DOUT + DOUT + t];
  __syncthreads();
  for (int off = 64; off > 0; off >>= 1) {
    if (t < off) { red0[t] += red0[t + off]; red1[t] += red1[t + off]; }
    __syncthreads();
  }
  if (t == 0) { sbuf[row] = red0[0]; dbuf[row] = red1[0]; }
}

// ---------------------------------------------------------------------------
// Stage 3: alpha[b,hd,i,:] = softmax_j( mask(lrelu(s_i + d_j)) )
// grid (NN, NH, BB), block 256, 8 j's per thread held in registers
// ---------------------------------------------------------------------------
__global__ void __launch_bounds__(256)
k_softmax(const float* __restrict__ sbuf, const float* __restrict__ dbuf,
          const int* __restrict__ adj, float* __restrict__ alpha) {
  const int i = blockIdx.x, hd = blockIdx.y, b = blockIdx.z;
  const int t = threadIdx.x;
  __shared__ float red[256];
  const size_t rowBH = ((size_t)b * NH + hd) * NN;
  const float sv = sbuf[rowBH + i];
  const float* dp = dbuf + rowBH;
  const int* ap = adj + ((size_t)b * NN + i) * NN;
  float e[8];
#pragma unroll
  for (int q = 0; q < 8; ++q) {
    const int j = t + q * 256;
    float x = sv + dp[j];
    x = x > 0.f ? x : NEG_SLOPE * x;
    e[q] = (ap[j] == 0) ? -__builtin_inff() : x;
  }
  float m = e[0];
#pragma unroll
  for (int q = 1; q < 8; ++q) m = fmaxf(m, e[q]);
  red[t] = m;
  __syncthreads();
  for (int off = 128; off > 0; off >>= 1) {
    if (t < off) red[t] = fmaxf(red[t], red[t + off]);
    __syncthreads();
  }
  m = red[0];
  __syncthreads();
  float sum = 0.f;
#pragma unroll
  for (int q = 0; q < 8; ++q) { e[q] = __expf(e[q] - m); sum += e[q]; }
  red[t] = sum;
  __syncthreads();
  for (int off = 128; off > 0; off >>= 1) {
    if (t < off) red[t] += red[t + off];
    __syncthreads();
  }
  const float inv = 1.f / red[0];
  float* outp = alpha + (rowBH + i) * NN;
#pragma unroll
  for (int q = 0; q < 8; ++q) outp[t + q * 256] = e[q] * inv;
}

// ---------------------------------------------------------------------------
// Stage 4: cat[b,n, hd*128+o] = sum_m alpha[b,hd,n,m] * h[b,hd,m,o]
// grid (NN/32, NH, BB), block 128; K = 2048 in steps of 4
// ---------------------------------------------------------------------------
__global__ void __launch_bounds__(128)
k_av(const float* __restrict__ alpha, const float* __restrict__ h,
     float* __restrict__ cat) {
  const int tm = blockIdx.x, hd = blockIdx.y, b = blockIdx.z;
  const int wave = threadIdx.x >> 5;
  const int lane = threadIdx.x & 31;
  const int half = lane >> 4;
  const int l16  = lane & 15;
  const int r0 = tm * 32;
  const int c0 = wave * 32;
  const size_t bh = (size_t)b * NH + hd;

  const float* A  = alpha + (bh * NN + r0) * NN;
  const float* Bm = h + bh * NN * DOUT + c0;
  v8f acc[2][2] = {};
  gemm_tile32<NN>(A, NN, Bm, DOUT, acc, half, l16);

#pragma unroll
  for (int i = 0; i < 2; ++i)
#pragma unroll
    for (int j = 0; j < 2; ++j)
#pragma unroll
      for (int v = 0; v < 8; ++v) {
        const int row = r0 + 16 * i + v + 8 * half;
        cat[((size_t)b * NN + row) * (NH * DOUT) + hd * DOUT + c0 + 16 * j +
            l16] = acc[i][j][v];
      }
}

// ---------------------------------------------------------------------------
// Stage 5: out[r,:] = cat[r,:] @ Wp + bias   (r over B*N = 8192 rows, K=512)
// ---------------------------------------------------------------------------
__global__ void __launch_bounds__(128)
k_outproj(const float* __restrict__ cat, const float* __restrict__ Wp,
          const float* __restrict__ bias, float* __restrict__ out) {
  const int tm = blockIdx.x;  // 0..255
  const int wave = threadIdx.x >> 5;
  const int lane = threadIdx.x & 31;
  const int half = lane >> 4;
  const int l16  = lane & 15;
  const int r0 = tm * 32;
  const int c0 = wave * 32;
  const int K = NH * DOUT;  // 512

  const float* A  = cat + (size_t)r0 * K;
  const float* Bm = Wp + c0;
  v8f acc[2][2] = {};
  gemm_tile32<NH * DOUT>(A, K, Bm, DOUT, acc, half, l16);

#pragma unroll
  for (int i = 0; i < 2; ++i)
#pragma unroll
    for (int j = 0; j < 2; ++j) {
      const float bsc = bias[c0 + 16 * j + l16];
#pragma unroll
      for (int v = 0; v < 8; ++v) {
        const int row = r0 + 16 * i + v + 8 * half;
        out[(size_t)row * DOUT + c0 + 16 * j + l16] = acc[i][j][v] + bsc;
      }
    }
}

// ---------------------------------------------------------------------------
extern "C" void kernel_launch(void* const* d_in, const int* in_sizes, int n_in,
                              void* d_out, int out_size, void* d_ws,
                              size_t ws_size, hipStream_t stream) {
  (void)in_sizes; (void)n_in; (void)out_size; (void)ws_size;
  const float* X    = (const float*)d_in[0];  // [B,N,128]
  const int*   adj  = (const int*)d_in[1];    // [B,N,N]
  const float* W    = (const float*)d_in[2];  // [H,128,128]
  const float* avec = (const float*)d_in[3];  // [H,256]
  const float* Wo   = (const float*)d_in[4];  // [512,128]
  const float* bo   = (const float*)d_in[5];  // [128]

  float* out   = (float*)d_out;                                  // [B,N,128]
  float* alpha = (float*)d_out + (size_t)BB * NN * DOUT;         // [B,H,N,N]

  float* ws   = (float*)d_ws;
  float* h    = ws;                                   // B*H*N*D   = 4,194,304
  float* sbuf = h + (size_t)BB * NH * NN * DOUT;      // B*H*N     =    32,768
  float* dbuf = sbuf + (size_t)BB * NH * NN;          // B*H*N     =    32,768
  float* cat  = dbuf + (size_t)BB * NH * NN;          // B*N*H*D   = 4,194,304

  k_proj<<<dim3(NN / 32, NH, BB), 128, 0, stream>>>(X, W, h);
  k_score<<<dim3(BB * NH * NN), 128, 0, stream>>>(h, avec, sbuf, dbuf);
  k_softmax<<<dim3(NN, NH, BB), 256, 0, stream>>>(sbuf, dbuf, adj, alpha);
  k_av<<<dim3(NN / 32, NH, BB), 128, 0, stream>>>(alpha, h, cat);
  k_outproj<<<dim3((BB * NN) / 32), 128, 0, stream>>>(cat, Wo, bo, out);
}